// Net_73220602462645
// MI455X (gfx1250) — compile-verified
//
#include <hip/hip_runtime.h>
#include <math.h>

typedef float v2f __attribute__((ext_vector_type(2)));
typedef float v8f __attribute__((ext_vector_type(8)));

#define FEAT_IN  128
#define FEAT_MID 16
#define FEAT_OUT 2

// ---- init: deg=1 (self-loop), acc1=0, acc2=0 -------------------------------
__global__ void init_ws(float* __restrict__ deg, float* __restrict__ acc1,
                        float* __restrict__ acc2, int n_nodes) {
    int tid = blockIdx.x * blockDim.x + threadIdx.x;
    if (tid < n_nodes * FEAT_MID) acc1[tid] = 0.0f;
    if (tid < n_nodes * FEAT_OUT) acc2[tid] = 0.0f;
    if (tid < n_nodes)            deg[tid]  = 1.0f;   // self-loop contribution
}

// ---- degree over dst -------------------------------------------------------
__global__ void degree_kernel(const int* __restrict__ dst, float* __restrict__ deg,
                              int n_edges) {
    int e = blockIdx.x * blockDim.x + threadIdx.x;
    if (e < n_edges) atomicAdd(&deg[dst[e]], 1.0f);
}

// ---- dinv = rsqrt(deg) in place -------------------------------------------
__global__ void dinv_kernel(float* __restrict__ deg, int n_nodes) {
    int i = blockIdx.x * blockDim.x + threadIdx.x;
    if (i < n_nodes) {
        float d = deg[i];
        deg[i] = (d > 0.0f) ? rsqrtf(d) : 0.0f;
    }
}

// ---- h = x @ W1 via V_WMMA_F32_16X16X4_F32 ---------------------------------
// One wave per 16-row node tile. 32 K-steps of 16x16x4 f32 WMMA.
__global__ void __launch_bounds__(256)
gemm1_wmma(const float* __restrict__ x, const float* __restrict__ W1,
           float* __restrict__ h, int n_tiles, int n_nodes) {
    int lane = threadIdx.x & 31;
    int wave = threadIdx.x >> 5;
    int tile = blockIdx.x * (blockDim.x >> 5) + wave;
    if (tile >= n_tiles) return;                 // wave-uniform: EXEC stays all-1s

    int col = lane & 15;                         // A: M row / B,D: N column
    int kh  = (lane >> 4) << 1;                  // K sub-offset: 0 or 2

    int rowA = tile * 16 + col;
    if (rowA > n_nodes - 1) rowA = n_nodes - 1;  // clamp (exact for N=100000)
    const float* xrow = x + (size_t)rowA * FEAT_IN;

    v8f acc = {};
    for (int k0 = 0; k0 < FEAT_IN; k0 += 4) {
        v2f a, b;
        a.x = xrow[k0 + kh];
        a.y = xrow[k0 + kh + 1];
        b.x = W1[(k0 + kh)     * FEAT_MID + col];
        b.y = W1[(k0 + kh + 1) * FEAT_MID + col];
        acc = __builtin_amdgcn_wmma_f32_16x16x4_f32(
            /*neg_a=*/false, a, /*neg_b=*/false, b,
            /*c_mod=*/(short)0, acc, /*reuse_a=*/false, /*reuse_b=*/false);
    }

    int mbase = (lane >> 4) * 8;                 // D layout: vgpr r -> row mbase+r
#pragma unroll
    for (int r = 0; r < 8; r++) {
        int row = tile * 16 + mbase + r;
        if (row < n_nodes) h[(size_t)row * FEAT_MID + col] = acc[r];
    }
}

// ---- layer-1 edge scatter: 16 lanes per edge (coalesced 64B gather) --------
__global__ void edge1_kernel(const int* __restrict__ src, const int* __restrict__ dst,
                             const float* __restrict__ dinv, const float* __restrict__ h,
                             float* __restrict__ acc1, int n_edges) {
    long long tid = (long long)blockIdx.x * blockDim.x + threadIdx.x;
    if (tid >= (long long)n_edges * FEAT_MID) return;
    int e = (int)(tid >> 4);
    int f = (int)(tid & 15);
    int s = src[e], d = dst[e];
    float norm = dinv[s] * dinv[d];
    atomicAdd(&acc1[d * FEAT_MID + f], h[s * FEAT_MID + f] * norm);
}

// ---- add self-loop term + bias, ReLU (in place over acc1) ------------------
__global__ void relu_bias1(const float* __restrict__ h, const float* __restrict__ dinv,
                           const float* __restrict__ b1, float* __restrict__ acc1,
                           int n_nodes) {
    int tid = blockIdx.x * blockDim.x + threadIdx.x;
    if (tid >= n_nodes * FEAT_MID) return;
    int node = tid >> 4, f = tid & 15;
    float di = dinv[node];
    float v = acc1[tid] + h[tid] * di * di + b1[f];
    acc1[tid] = fmaxf(v, 0.0f);
}

// ---- h2 = h1 @ W2 (16 -> 2): too skinny for WMMA, plain FMA ----------------
__global__ void gemm2_kernel(const float* __restrict__ h1, const float* __restrict__ W2,
                             float* __restrict__ h2, int n_nodes) {
    int i = blockIdx.x * blockDim.x + threadIdx.x;
    if (i >= n_nodes) return;
    float s0 = 0.0f, s1 = 0.0f;
#pragma unroll
    for (int k = 0; k < FEAT_MID; k++) {
        float v = h1[i * FEAT_MID + k];
        s0 = fmaf(v, W2[k * 2 + 0], s0);
        s1 = fmaf(v, W2[k * 2 + 1], s1);
    }
    h2[i * 2 + 0] = s0;
    h2[i * 2 + 1] = s1;
}

// ---- layer-2 edge scatter: 2 lanes per edge --------------------------------
__global__ void edge2_kernel(const int* __restrict__ src, const int* __restrict__ dst,
                             const float* __restrict__ dinv, const float* __restrict__ h2,
                             float* __restrict__ acc2, int n_edges) {
    long long tid = (long long)blockIdx.x * blockDim.x + threadIdx.x;
    if (tid >= (long long)n_edges * FEAT_OUT) return;
    int e = (int)(tid >> 1);
    int c = (int)(tid & 1);
    int s = src[e], d = dst[e];
    float norm = dinv[s] * dinv[d];
    atomicAdd(&acc2[d * FEAT_OUT + c], h2[s * FEAT_OUT + c] * norm);
}

// ---- self-loop + bias + log_softmax over 2 classes -------------------------
__global__ void final_kernel(const float* __restrict__ acc2, const float* __restrict__ h2,
                             const float* __restrict__ dinv, const float* __restrict__ b2,
                             float* __restrict__ out, int n_nodes) {
    int i = blockIdx.x * blockDim.x + threadIdx.x;
    if (i >= n_nodes) return;
    float di = dinv[i];
    float sc = di * di;
    float v0 = acc2[i * 2 + 0] + h2[i * 2 + 0] * sc + b2[0];
    float v1 = acc2[i * 2 + 1] + h2[i * 2 + 1] * sc + b2[1];
    float m  = fmaxf(v0, v1);
    float l  = m + logf(expf(v0 - m) + expf(v1 - m));
    out[i * 2 + 0] = v0 - l;
    out[i * 2 + 1] = v1 - l;
}

extern "C" void kernel_launch(void* const* d_in, const int* in_sizes, int n_in,
                              void* d_out, int out_size, void* d_ws, size_t ws_size,
                              hipStream_t stream) {
    const float* x  = (const float*)d_in[0];
    const int*   ei = (const int*)d_in[1];
    const float* W1 = (const float*)d_in[2];
    const float* b1 = (const float*)d_in[3];
    const float* W2 = (const float*)d_in[4];
    const float* b2 = (const float*)d_in[5];
    float* out = (float*)d_out;

    int n_nodes = in_sizes[0] / FEAT_IN;   // 100000
    int n_edges = in_sizes[1] / 2;         // 3200000
    const int* src = ei;
    const int* dst = ei + n_edges;

    // workspace layout (floats): deg/dinv | h[N,16] | acc1/h1[N,16] | h2[N,2] | acc2[N,2]
    float* ws   = (float*)d_ws;
    float* deg  = ws;
    float* h    = deg  + n_nodes;
    float* acc1 = h    + (size_t)n_nodes * FEAT_MID;
    float* h2   = acc1 + (size_t)n_nodes * FEAT_MID;
    float* acc2 = h2   + (size_t)n_nodes * FEAT_OUT;

    const int B = 256;
    int n16 = n_nodes * FEAT_MID;

    init_ws<<<(n16 + B - 1) / B, B, 0, stream>>>(deg, acc1, acc2, n_nodes);
    degree_kernel<<<(n_edges + B - 1) / B, B, 0, stream>>>(dst, deg, n_edges);
    dinv_kernel<<<(n_nodes + B - 1) / B, B, 0, stream>>>(deg, n_nodes);

    int n_tiles = (n_nodes + 15) / 16;     // 6250
    int wpb = B / 32;                      // waves per block
    gemm1_wmma<<<(n_tiles + wpb - 1) / wpb, B, 0, stream>>>(x, W1, h, n_tiles, n_nodes);

    long long e16 = (long long)n_edges * FEAT_MID;
    edge1_kernel<<<(int)((e16 + B - 1) / B), B, 0, stream>>>(src, dst, deg, h, acc1, n_edges);
    relu_bias1<<<(n16 + B - 1) / B, B, 0, stream>>>(h, deg, b1, acc1, n_nodes);

    gemm2_kernel<<<(n_nodes + B - 1) / B, B, 0, stream>>>(acc1, W2, h2, n_nodes);

    long long e2 = (long long)n_edges * FEAT_OUT;
    edge2_kernel<<<(int)((e2 + B - 1) / B), B, 0, stream>>>(src, dst, deg, h2, acc2, n_edges);

    final_kernel<<<(n_nodes + B - 1) / B, B, 0, stream>>>(acc2, h2, deg, b2, out, n_nodes);
}